// MultimodalGNN_17068200034898
// MI455X (gfx1250) — compile-verified
//
#include <hip/hip_runtime.h>

typedef float v2f __attribute__((ext_vector_type(2)));
typedef float v8f __attribute__((ext_vector_type(8)));

#define N_NODES 100000
#define IN_DIM  384
#define HID     64

// ---------------- degree / normalization ----------------

__global__ void k_zero_deg(unsigned int* __restrict__ deg) {
    int i = blockIdx.x * blockDim.x + threadIdx.x;
    if (i < N_NODES) deg[i] = 0u;
}

__global__ void k_deg_count(const int* __restrict__ dst, int E,
                            unsigned int* __restrict__ deg) {
    int e = blockIdx.x * blockDim.x + threadIdx.x;
    if (e < E) atomicAdd(&deg[dst[e]], 1u);
}

__global__ void k_dinv(const unsigned int* __restrict__ deg,
                       float* __restrict__ dinv) {
    int i = blockIdx.x * blockDim.x + threadIdx.x;
    if (i < N_NODES) dinv[i] = rsqrtf((float)deg[i] + 1.0f);  // +1 self-loop
}

// ---------------- H = X @ W1 via fp32 WMMA 16x16x4 ----------------
// One wave computes a 16x16 output tile; 4 waves/block cover the 64 cols.
// A layout (16x4 f32): lanes 0-15 hold M=0..15; VGPR0 = K=(0|2), VGPR1 = K=(1|3)
// selected by lane-half.  B layout (4x16): N in lanes, K across VGPR+half.
// C/D layout (16x16): N in lanes, VGPR r -> M=r (+8 for upper lane half).
__global__ __launch_bounds__(128) void k_gemm1(const float* __restrict__ X,
                                               const float* __restrict__ W1,
                                               float* __restrict__ H) {
    const int lane = threadIdx.x & 31;
    const int wave = threadIdx.x >> 5;
    const int half = lane >> 4;     // 0 or 1
    const int lr   = lane & 15;
    const int rowBase = blockIdx.x * 16;
    const int colBase = wave * 16;

    v8f c = {};
    const float* xrow = X + (size_t)(rowBase + lr) * IN_DIM;
    for (int k0 = 0; k0 < IN_DIM; k0 += 4) {
        v2f a, b;
        const float* ap = xrow + k0 + 2 * half;
        a[0] = ap[0];
        a[1] = ap[1];
        const float* bp = W1 + (size_t)(k0 + 2 * half) * HID + colBase + lr;
        b[0] = bp[0];
        b[1] = bp[HID];
        c = __builtin_amdgcn_wmma_f32_16x16x4_f32(
                /*neg_a=*/false, a, /*neg_b=*/false, b,
                /*c_mod=*/(short)0, c, /*reuse_a=*/false, /*reuse_b=*/false);
    }
    float* hrow = H + (size_t)(rowBase + 8 * half) * HID + colBase + lr;
#pragma unroll
    for (int r = 0; r < 8; ++r)
        hrow[(size_t)r * HID] = c[r];
}

// ---------------- layer-1 aggregation ----------------

// agg[i][f] = H[i][f] * dinv[i]^2   (self-loop contribution, also zero-inits agg)
__global__ void k_selfloop1(const float* __restrict__ H,
                            const float* __restrict__ dinv,
                            float* __restrict__ agg) {
    int idx = blockIdx.x * blockDim.x + threadIdx.x;
    if (idx < N_NODES * HID) {
        float di = dinv[idx >> 6];
        agg[idx] = H[idx] * di * di;
    }
}

// 64 consecutive threads per edge: coalesced gather of H[src], L2-resident
// fp32 atomic scatter into agg[dst].
__global__ void k_scatter1(const int* __restrict__ src,
                           const int* __restrict__ dst, int E,
                           const float* __restrict__ H,
                           const float* __restrict__ dinv,
                           float* __restrict__ agg) {
    int i = blockIdx.x * blockDim.x + threadIdx.x;
    int e = i >> 6;
    if (e < E) {
        int f = i & 63;
        int s = src[e], d = dst[e];
        float w = dinv[s] * dinv[d];
        atomicAdd(&agg[(size_t)d * HID + f], H[(size_t)s * HID + f] * w);
    }
}

// ---------------- layer 2: z[i] = relu(agg[i]+b1) . W2 ----------------
// One wave per row; lane handles feats (lane, lane+32); wave32 shuffle reduce.
__global__ __launch_bounds__(256) void k_dot2(const float* __restrict__ agg,
                                              const float* __restrict__ b1,
                                              const float* __restrict__ W2,
                                              float* __restrict__ z) {
    int row  = (blockIdx.x * blockDim.x + threadIdx.x) >> 5;
    int lane = threadIdx.x & 31;
    if (row < N_NODES) {
        const float* a = agg + (size_t)row * HID;
        float v0 = a[lane] + b1[lane];
        float v1 = a[lane + 32] + b1[lane + 32];
        v0 = v0 > 0.0f ? v0 : 0.0f;
        v1 = v1 > 0.0f ? v1 : 0.0f;
        float s = v0 * W2[lane] + v1 * W2[lane + 32];
#pragma unroll
        for (int off = 16; off > 0; off >>= 1)
            s += __shfl_down(s, off, 32);
        if (lane == 0) z[row] = s;
    }
}

// out[i] = z[i]*dinv[i]^2 + b2   (self-loop + bias, zero-inits out)
__global__ void k_out_init(const float* __restrict__ z,
                           const float* __restrict__ dinv,
                           const float* __restrict__ b2,
                           float* __restrict__ out) {
    int i = blockIdx.x * blockDim.x + threadIdx.x;
    if (i < N_NODES) {
        float di = dinv[i];
        out[i] = z[i] * di * di + b2[0];
    }
}

__global__ void k_scatter2(const int* __restrict__ src,
                           const int* __restrict__ dst, int E,
                           const float* __restrict__ z,
                           const float* __restrict__ dinv,
                           float* __restrict__ out) {
    int e = blockIdx.x * blockDim.x + threadIdx.x;
    if (e < E) {
        int s = src[e], d = dst[e];
        atomicAdd(&out[d], z[s] * dinv[s] * dinv[d]);
    }
}

// ---------------- launcher ----------------

extern "C" void kernel_launch(void* const* d_in, const int* in_sizes, int n_in,
                              void* d_out, int out_size, void* d_ws, size_t ws_size,
                              hipStream_t stream) {
    const float* X  = (const float*)d_in[0];
    const int*   EI = (const int*)d_in[1];   // edge_index, (2, E) flat
    const float* W1 = (const float*)d_in[2];
    const float* b1 = (const float*)d_in[3];
    const float* W2 = (const float*)d_in[4];
    const float* b2 = (const float*)d_in[5];
    float* out = (float*)d_out;

    const int E = in_sizes[1] / 2;
    const int* src = EI;
    const int* dst = EI + E;

    char* ws = (char*)d_ws;
    float* H    = (float*)ws;                                     // 25.6 MB
    float* agg  = (float*)(ws + (size_t)N_NODES * HID * 4);       // 25.6 MB
    float* dinv = (float*)(ws + (size_t)2 * N_NODES * HID * 4);   // 0.4 MB
    float* z    = dinv + N_NODES;                                 // 0.4 MB
    unsigned int* deg = (unsigned int*)(z + N_NODES);             // 0.4 MB

    const int B = 256;
    k_zero_deg <<<(N_NODES + B - 1) / B, B, 0, stream>>>(deg);
    k_deg_count<<<(E + B - 1) / B, B, 0, stream>>>(dst, E, deg);
    k_dinv     <<<(N_NODES + B - 1) / B, B, 0, stream>>>(deg, dinv);

    k_gemm1    <<<N_NODES / 16, 128, 0, stream>>>(X, W1, H);      // 100000 % 16 == 0

    k_selfloop1<<<(N_NODES * HID + B - 1) / B, B, 0, stream>>>(H, dinv, agg);
    long long work1 = (long long)E * 64;
    k_scatter1 <<<(int)((work1 + B - 1) / B), B, 0, stream>>>(src, dst, E, H, dinv, agg);

    k_dot2     <<<(N_NODES * 32 + B - 1) / B, B, 0, stream>>>(agg, b1, W2, z);
    k_out_init <<<(N_NODES + B - 1) / B, B, 0, stream>>>(z, dinv, b2, out);
    k_scatter2 <<<(E + B - 1) / B, B, 0, stream>>>(src, dst, E, z, dinv, out);
}